// RegularizedFMNet_4114578669826
// MI455X (gfx1250) — compile-verified
//
#include <hip/hip_runtime.h>
#include <hip/hip_bf16.h>

typedef __attribute__((ext_vector_type(2))) float v2f;
typedef __attribute__((ext_vector_type(8))) float v8f;

static constexpr int Bn = 8;      // batch
static constexpr int V  = 12000;  // vertices
static constexpr int C  = 256;    // feature channels
static constexpr int K  = 128;    // spectral basis
static constexpr float LMBDA = 100.0f;
static constexpr float EPSV  = 1e-6f;
static constexpr float TINY  = 1e-10f;

// ---------------------------------------------------------------------------
// Kernel 1: A = E (B,K,V) @ F (B,V,C) -> (B,K,C).
// One wave per 16(M) x 64(N) tile: 4 f32 accumulators, A-operand amortized
// across 4 WMMAs per K-step. blockDim = 256 (8 waves).
// grid = (32 tiles / 8, B, 2); z selects (Ex,Fx)->A vs (Ey,Fy)->Bm.
// ---------------------------------------------------------------------------
__global__ __launch_bounds__(256)
void fmnet_proj_gemm(const float* __restrict__ Ex, const float* __restrict__ Fx,
                     const float* __restrict__ Ey, const float* __restrict__ Fy,
                     float* __restrict__ outA, float* __restrict__ outB)
{
    const int b = blockIdx.y;
    const float* E; const float* F; float* O;
    if (blockIdx.z == 0) { E = Ex; F = Fx; O = outA; }
    else                 { E = Ey; F = Fy; O = outB; }

    const int wave = threadIdx.x >> 5;
    const int tile = blockIdx.x * 8 + wave;          // 0..31
    const int mt = tile >> 2;                        // K/16 tiles: 0..7
    const int nt = tile & 3;                         // C/64 tiles: 0..3
    const int m0 = mt * 16;
    const int n0 = nt * 64;

    const int lane = threadIdx.x & 31;
    const int row  = lane & 15;                      // M (for A) / N (for B)
    const int hi   = lane >> 4;                      // selects K-pair {0,1} vs {2,3}

    // A operand: E[b][m0+row][v + hi*2 + {0,1}] -> contiguous float2
    const float* Erow  = E + ((size_t)b * K + m0 + row) * V + (size_t)(hi * 2);
    // B operand: F[b][v + hi*2 + {0,1}][n0 + 16j + row] -> stride-C pair
    const float* Fbase = F + (size_t)b * V * C + n0 + row;

    v8f acc0 = {}, acc1 = {}, acc2 = {}, acc3 = {};
    for (int v = 0; v < V; v += 4) {
        const v2f a = *(const v2f*)(Erow + v);
        const size_t fr = (size_t)(v + hi * 2) * C;

        // prefetch the F stream ~32 rows ahead (lowers to global_prefetch_b8)
        __builtin_prefetch(Fbase + fr + (size_t)32 * C, 0, 1);

        v2f bb;
        bb.x = Fbase[fr +  0]; bb.y = Fbase[fr + C +  0];
        acc0 = __builtin_amdgcn_wmma_f32_16x16x4_f32(false, a, false, bb, (short)0, acc0, false, false);
        bb.x = Fbase[fr + 16]; bb.y = Fbase[fr + C + 16];
        acc1 = __builtin_amdgcn_wmma_f32_16x16x4_f32(false, a, false, bb, (short)0, acc1, false, false);
        bb.x = Fbase[fr + 32]; bb.y = Fbase[fr + C + 32];
        acc2 = __builtin_amdgcn_wmma_f32_16x16x4_f32(false, a, false, bb, (short)0, acc2, false, false);
        bb.x = Fbase[fr + 48]; bb.y = Fbase[fr + C + 48];
        acc3 = __builtin_amdgcn_wmma_f32_16x16x4_f32(false, a, false, bb, (short)0, acc3, false, false);
    }

    // D layout: VGPR r holds (M = r + hi*8, N = row)
    float* Orow = O + ((size_t)b * K + m0 + hi * 8) * C + n0 + row;
#pragma unroll
    for (int r = 0; r < 8; ++r) {
        Orow[(size_t)r * C +  0] = acc0[r];
        Orow[(size_t)r * C + 16] = acc1[r];
        Orow[(size_t)r * C + 32] = acc2[r];
        Orow[(size_t)r * C + 48] = acc3[r];
    }
}

// ---------------------------------------------------------------------------
// Kernel 2: out (B,K,K) = X (B,K,C) @ A^T.  Both operands contiguous in C.
// One wave per 16(M) x 64(N) tile.  blockDim = 256 (8 waves).
// grid = (16 tiles / 8, B, 2); z=0: X=A (AAt), z=1: X=Bm (BAt)
// ---------------------------------------------------------------------------
__global__ __launch_bounds__(256)
void fmnet_gram_gemm(const float* __restrict__ Amat, const float* __restrict__ Bmat,
                     float* __restrict__ outAA, float* __restrict__ outBA)
{
    const int b = blockIdx.y;
    const float* X; float* O;
    if (blockIdx.z == 0) { X = Amat; O = outAA; }
    else                 { X = Bmat; O = outBA; }

    const int wave = threadIdx.x >> 5;
    const int tile = blockIdx.x * 8 + wave;          // 0..15
    const int mt = tile >> 1;                        // 0..7
    const int nt = tile & 1;                         // 0..1
    const int m0 = mt * 16;
    const int n0 = nt * 64;

    const int lane = threadIdx.x & 31;
    const int row  = lane & 15;
    const int hi   = lane >> 4;

    const float* Xrow = X    + ((size_t)b * K + m0 + row) * C + (size_t)(hi * 2);
    const float* Ar0  = Amat + ((size_t)b * K + n0 +  0 + row) * C + (size_t)(hi * 2);
    const float* Ar1  = Amat + ((size_t)b * K + n0 + 16 + row) * C + (size_t)(hi * 2);
    const float* Ar2  = Amat + ((size_t)b * K + n0 + 32 + row) * C + (size_t)(hi * 2);
    const float* Ar3  = Amat + ((size_t)b * K + n0 + 48 + row) * C + (size_t)(hi * 2);

    v8f acc0 = {}, acc1 = {}, acc2 = {}, acc3 = {};
#pragma unroll 4
    for (int c = 0; c < C; c += 4) {
        const v2f a = *(const v2f*)(Xrow + c);
        acc0 = __builtin_amdgcn_wmma_f32_16x16x4_f32(false, a, false, *(const v2f*)(Ar0 + c), (short)0, acc0, false, false);
        acc1 = __builtin_amdgcn_wmma_f32_16x16x4_f32(false, a, false, *(const v2f*)(Ar1 + c), (short)0, acc1, false, false);
        acc2 = __builtin_amdgcn_wmma_f32_16x16x4_f32(false, a, false, *(const v2f*)(Ar2 + c), (short)0, acc2, false, false);
        acc3 = __builtin_amdgcn_wmma_f32_16x16x4_f32(false, a, false, *(const v2f*)(Ar3 + c), (short)0, acc3, false, false);
    }

    float* Orow = O + ((size_t)b * K + m0 + hi * 8) * K + n0 + row;
#pragma unroll
    for (int r = 0; r < 8; ++r) {
        Orow[(size_t)r * K +  0] = acc0[r];
        Orow[(size_t)r * K + 16] = acc1[r];
        Orow[(size_t)r * K + 32] = acc2[r];
        Orow[(size_t)r * K + 48] = acc3[r];
    }
}

// ---------------------------------------------------------------------------
// Kernel 3: per-batch eigenvalue rescale e -> (e/s)^0.5 with s = rowwise max.
// grid = B, blockDim = K (128)
// ---------------------------------------------------------------------------
__global__ __launch_bounds__(128)
void fmnet_scale_evals(const float* __restrict__ ex, const float* __restrict__ ey,
                       float* __restrict__ e1p, float* __restrict__ e2p)
{
    const int b = blockIdx.x;
    const int t = threadIdx.x;

    float e1 = ex[b * K + t];
    float e2 = ey[b * K + t];
    if (fabsf(e1) < EPSV) e1 = TINY;
    if (fabsf(e2) < EPSV) e2 = TINY;
    e1 = fmaxf(e1, TINY);
    e2 = fmaxf(e2, TINY);

    __shared__ float red[128];
    red[t] = fmaxf(e1, e2);
    __syncthreads();
    for (int s = 64; s > 0; s >>= 1) {
        if (t < s) red[t] = fmaxf(red[t], red[t + s]);
        __syncthreads();
    }
    const float s = red[0];

    float p1 = sqrtf(e1 / s);   // gamma = 0.5
    float p2 = sqrtf(e2 / s);
    if (!(p1 == p1)) p1 = TINY; p1 = fminf(p1, 1.0e6f);
    if (!(p2 == p2)) p2 = TINY; p2 = fminf(p2, 1.0e6f);

    e1p[b * K + t] = p1;
    e2p[b * K + t] = p2;
}

// ---------------------------------------------------------------------------
// Kernel 4: per (b,i) solve (AAt[b] + LMBDA*diag(D[b,i,:])) x = BAt[b,i,:]
// grid = (K, B), blockDim = K (128).  Gaussian elimination in padded LDS.
// ---------------------------------------------------------------------------
__global__ __launch_bounds__(128)
void fmnet_solve(const float* __restrict__ AAt, const float* __restrict__ BAt,
                 const float* __restrict__ e1p, const float* __restrict__ e2p,
                 float* __restrict__ out)
{
    const int b = blockIdx.y;
    const int i = blockIdx.x;
    const int t = threadIdx.x;

    __shared__ float Ms[K][K + 1];     // 129-stride: conflict-free column access
    __shared__ float rhs[K];

    const float* Ab = AAt + (size_t)b * K * K;
    for (int idx = t; idx < K * K; idx += K) {   // coalesced
        const int r = idx >> 7;
        const int c = idx & (K - 1);
        Ms[r][c] = Ab[idx];
    }
    rhs[t] = BAt[((size_t)b * K + i) * K + t];

    // mask D[b][i][t]: g1 varies along j (=t), g2 fixed per system (=i)
    const float g1 = e1p[b * K + t];
    const float g2 = e2p[b * K + i];
    const float d1 = fmaxf(g1 * g1 + 1.0f, TINY);
    const float d2 = fmaxf(g2 * g2 + 1.0f, TINY);
    const float Mre = g2 / d2 - g1 / d1;
    const float Mim = 1.0f / d2 - 1.0f / d1;
    float mask = Mre * Mre + Mim * Mim;
    if (!(mask == mask)) mask = 0.0f;
    mask = fminf(mask, 1.0e6f);

    __syncthreads();
    Ms[t][t] += LMBDA * mask;

    // forward elimination (no pivoting; matrix is Gram + positive diagonal)
    for (int p = 0; p < K - 1; ++p) {
        __syncthreads();
        if (t > p) {
            const float f = Ms[t][p] / Ms[p][p];
            for (int c = p + 1; c < K; ++c)
                Ms[t][c] -= f * Ms[p][c];
            rhs[t] -= f * rhs[p];
        }
    }

    // back substitution
    for (int p = K - 1; p >= 0; --p) {
        __syncthreads();
        if (t == p) rhs[p] /= Ms[p][p];
        __syncthreads();
        if (t < p) rhs[t] -= Ms[t][p] * rhs[p];
    }
    __syncthreads();

    out[((size_t)b * K + i) * K + t] = rhs[t];
}

// ---------------------------------------------------------------------------
extern "C" void kernel_launch(void* const* d_in, const int* in_sizes, int n_in,
                              void* d_out, int out_size, void* d_ws, size_t ws_size,
                              hipStream_t stream)
{
    const float* feat_x  = (const float*)d_in[0];
    const float* feat_y  = (const float*)d_in[1];
    const float* evals_x = (const float*)d_in[2];
    const float* evals_y = (const float*)d_in[3];
    const float* evtx    = (const float*)d_in[4];
    const float* evty    = (const float*)d_in[5];
    float* Cxy = (float*)d_out;

    // workspace layout
    float* Amat = (float*)d_ws;                        // B*K*C
    float* Bmat = Amat + (size_t)Bn * K * C;           // B*K*C
    float* AAt  = Bmat + (size_t)Bn * K * C;           // B*K*K
    float* BAt  = AAt  + (size_t)Bn * K * K;           // B*K*K
    float* e1p  = BAt  + (size_t)Bn * K * K;           // B*K
    float* e2p  = e1p  + (size_t)Bn * K;               // B*K

    // 1) spectral projections: A = Ex@Fx, Bm = Ey@Fy
    {
        dim3 grid((K / 16) * (C / 64) / 8, Bn, 2);     // (4, 8, 2)
        fmnet_proj_gemm<<<grid, 256, 0, stream>>>(evtx, feat_x, evty, feat_y, Amat, Bmat);
    }
    // 2) Gram matrices: AAt = A@A^T, BAt = Bm@A^T
    {
        dim3 grid((K / 16) * (K / 64) / 8, Bn, 2);     // (2, 8, 2)
        fmnet_gram_gemm<<<grid, 256, 0, stream>>>(Amat, Bmat, AAt, BAt);
    }
    // 3) eigenvalue rescale
    fmnet_scale_evals<<<Bn, K, 0, stream>>>(evals_x, evals_y, e1p, e2p);
    // 4) 1024 regularized solves
    {
        dim3 grid(K, Bn);
        fmnet_solve<<<grid, K, 0, stream>>>(AAt, BAt, e1p, e2p, Cxy);
    }
}